// QuadConvLayer_24180665877005
// MI455X (gfx1250) — compile-verified
//
#include <hip/hip_runtime.h>

typedef __attribute__((ext_vector_type(2))) float v2f;
typedef __attribute__((ext_vector_type(8))) float v8f;

#define B_     64
#define CIN_   8
#define COUT_  8
#define N_     600
#define M_     300
#define NWIN   48          // n-window per 16-row m-tile (covers all nonzero bump terms)
#define MTILES 19          // ceil(300/16)

__device__ __forceinline__ v8f wmma4(v2f a, v2f b, v8f c) {
  // D = A(16x4 f32) * B(4x16 f32) + C(16x16 f32)
  return __builtin_amdgcn_wmma_f32_16x16x4_f32(false, a, false, b, (short)0, c,
                                               false, false);
}

// kernel[co,ci,m,n] value = MLP(d) * bump(d) * quad_weight[n]
__device__ __forceinline__ float eval_kernel(float d, const float* __restrict__ w1,
                                             const float* __restrict__ w2,
                                             const float* __restrict__ w3,
                                             const float* __restrict__ w4,
                                             float qw) {
  const float decay     = 506250000.0f;       // (600/4)^4
  const float inv_decay = 1.0f / decay;
  float d2 = d * d;
  float b4 = d2 * d2;
  if (!(b4 <= inv_decay)) return 0.0f;        // bump exactly zero -> whole term zero
  float t    = 1.0f - decay * b4;             // >= 0; t==0 -> -inf -> exp -> 0
  float bump = 2.7182818284590452f * __expf(-1.0f / t);
  // 1 -> 4 -> 8 -> 4 -> 1 sin-MLP, no bias
  float h1[4], h2[8], h3[4];
#pragma unroll
  for (int j = 0; j < 4; ++j) h1[j] = __sinf(w1[j] * d);
#pragma unroll
  for (int g = 0; g < 8; ++g) {
    float a = 0.0f;
#pragma unroll
    for (int j = 0; j < 4; ++j) a += w2[g * 4 + j] * h1[j];
    h2[g] = __sinf(a);
  }
#pragma unroll
  for (int f = 0; f < 4; ++f) {
    float a = 0.0f;
#pragma unroll
    for (int g = 0; g < 8; ++g) a += w3[f * 8 + g] * h2[g];
    h3[f] = __sinf(a);
  }
  float wv = 0.0f;
#pragma unroll
  for (int f = 0; f < 4; ++f) wv += w4[f] * h3[f];
  return wv * bump * qw;
}

__global__ __launch_bounds__(256) void quadconv_wmma_kernel(
    const float* __restrict__ features,     // [64,8,600]
    const float* __restrict__ output_locs,  // [300]
    const float* __restrict__ quad_nodes,   // [600]
    const float* __restrict__ quad_weights, // [600]
    const float* __restrict__ W1,           // [64,4,1]
    const float* __restrict__ W2,           // [64,8,4]
    const float* __restrict__ W3,           // [64,4,8]
    const float* __restrict__ W4,           // [64,1,4]
    float* __restrict__ out) {              // [64,8,300]
  __shared__ float s_w1[CIN_ * 4];
  __shared__ float s_w2[CIN_ * 32];
  __shared__ float s_w3[CIN_ * 32];
  __shared__ float s_w4[CIN_ * 4];
  __shared__ float s_nodes[NWIN];
  __shared__ float s_qw[NWIN];
  __shared__ float s_red[CIN_][16 * 64];    // per-wave partial tiles, 32 KB

  const int tid = threadIdx.x;
  const int bx  = blockIdx.x;
  const int co  = bx % COUT_;
  const int mt  = bx / COUT_;
  const int m0  = mt * 16;

  // n-window start: bump support is |loc[m]-node[n]| <= 1/150  ->  ~4 nodes each side
  int n_start = (int)floorf((599.0f * (float)m0) / 299.0f - 4.01f);
  if (n_start < 0) n_start = 0;

  // Stage node/weight window (OOB n -> sentinel node => bump==0, qw=0)
  for (int i = tid; i < NWIN; i += 256) {
    int  n  = n_start + i;
    bool ok = (n < N_);
    s_nodes[i] = ok ? quad_nodes[n] : 1.0e9f;
    s_qw[i]    = ok ? quad_weights[n] : 0.0f;
  }
  // Stage this co's 8 tiny-MLP weight sets (e = co*CIN + ci)
  for (int i = tid; i < CIN_ * 72; i += 256) {
    int ci = i / 72, r = i % 72;
    int e  = co * CIN_ + ci;
    if (r < 4)       s_w1[ci * 4  + r]        = W1[e * 4  + r];
    else if (r < 36) s_w2[ci * 32 + (r - 4)]  = W2[e * 32 + (r - 4)];
    else if (r < 68) s_w3[ci * 32 + (r - 36)] = W3[e * 32 + (r - 36)];
    else             s_w4[ci * 4  + (r - 68)] = W4[e * 4  + (r - 68)];
  }
  __syncthreads();

  const int wave = tid >> 5;   // 8 waves: wave == ci
  const int lane = tid & 31;
  const int half = lane >> 4;  // K-half select (A/B VGPR layout)
  const int lm   = lane & 15;  // A row (m) / B col (b) within group
  const int ci   = wave;

  const int   m   = m0 + lm;
  const float loc = (m < M_) ? output_locs[m] : 2.0f;  // d>=1 => bump exactly 0

  const float* w1 = &s_w1[ci * 4];
  const float* w2 = &s_w2[ci * 32];
  const float* w3 = &s_w3[ci * 32];
  const float* w4 = &s_w4[ci * 4];

  v8f acc[4];
#pragma unroll
  for (int g = 0; g < 4; ++g) acc[g] = (v8f){};

  // K loop: 48 window n's in steps of 4 (one wmma K per step, 4 b-groups)
  for (int ks = 0; ks < NWIN / 4; ++ks) {
    const int i0 = ks * 4 + 2 * half;   // window-local K index for this lane-half
    const int i1 = i0 + 1;
    const int n0 = n_start + i0;
    const int n1 = n0 + 1;

    // A fragment: kernel values at (m, n0), (m, n1)
    v2f A;
    A[0] = eval_kernel(loc - s_nodes[i0], w1, w2, w3, w4, s_qw[i0]);
    A[1] = eval_kernel(loc - s_nodes[i1], w1, w2, w3, w4, s_qw[i1]);

    // B fragments: features[b, ci, n], 4 groups of 16 b's
#pragma unroll
    for (int g = 0; g < 4; ++g) {
      const int    b  = g * 16 + lm;
      const float* fb = features + (size_t)b * (CIN_ * N_) + ci * N_;
      v2f Bf;
      Bf[0] = (n0 < N_) ? fb[n0] : 0.0f;
      Bf[1] = (n1 < N_) ? fb[n1] : 0.0f;
      acc[g] = wmma4(A, Bf, acc[g]);
    }
  }

  // Spill per-wave (per-ci) partial 16x64 tiles to LDS
#pragma unroll
  for (int r = 0; r < 8; ++r) {
    const int ml = r + 8 * half;        // C/D layout: VGPR r, lane-half selects +8
#pragma unroll
    for (int g = 0; g < 4; ++g) s_red[wave][ml * 64 + g * 16 + lm] = acc[g][r];
  }
  __syncthreads();

  // Deterministic fixed-order reduction over ci, then store out[b, co, m]
  for (int idx = tid; idx < 16 * 64; idx += 256) {
    const int ml = idx >> 6;
    const int b  = idx & 63;
    float s = 0.0f;
#pragma unroll
    for (int w = 0; w < CIN_; ++w) s += s_red[w][ml * 64 + b];
    const int mm = m0 + ml;
    if (mm < M_) out[(size_t)b * (COUT_ * M_) + co * M_ + mm] = s;
  }
}

extern "C" void kernel_launch(void* const* d_in, const int* in_sizes, int n_in,
                              void* d_out, int out_size, void* d_ws, size_t ws_size,
                              hipStream_t stream) {
  const float* features     = (const float*)d_in[0];
  const float* output_locs  = (const float*)d_in[1];
  const float* quad_nodes   = (const float*)d_in[2];
  const float* quad_weights = (const float*)d_in[3];
  const float* W1           = (const float*)d_in[4];
  const float* W2           = (const float*)d_in[5];
  const float* W3           = (const float*)d_in[6];
  const float* W4           = (const float*)d_in[7];
  float*       out          = (float*)d_out;

  dim3 grid(COUT_ * MTILES);   // 152 workgroups: (co, m-tile)
  dim3 block(256);             // 8 wave32 waves: wave == ci
  quadconv_wmma_kernel<<<grid, block, 0, stream>>>(
      features, output_locs, quad_nodes, quad_weights, W1, W2, W3, W4, out);
}